// Repro_87402584474062
// MI455X (gfx1250) — compile-verified
//
#include <hip/hip_runtime.h>

typedef float v2f __attribute__((ext_vector_type(2)));
typedef float v8f __attribute__((ext_vector_type(8)));

#define BB 6     // batches
#define SS 256   // slots
#define DD 256   // feature dim
#define AA 12    // query rows

// out1[a,b,d] = sum_c q[a,b,c] * add[b,c,d]
//   add[b,c,d] = (c == slot[b] ? table[row[b]*DD + colid[d]] : bank[b,c,d]) + 0.975*ema[b,c,d]
// One wave per (batch b, 16-wide N tile): 6*16 = 96 tiles, K=256 stepped by 4
// with V_WMMA_F32_16X16X4_F32 (f32 A/B/C — exact dtype match with reference).
// All selects are branchless so EXEC stays all-ones around the WMMA ops.
__global__ __launch_bounds__(32)
void out1_wmma_kernel(const float* __restrict__ bank,
                      const float* __restrict__ table,
                      const int*   __restrict__ rows,    // primals_3 [6]
                      const int*   __restrict__ colid,   // primals_4 [256]
                      const int*   __restrict__ slots,   // primals_5 [6]
                      const float* __restrict__ ema,
                      const float* __restrict__ q,       // [12,6,256]
                      float*       __restrict__ out1) {  // [12,6,256]
    const int tile = blockIdx.x;          // 0..95
    const int b    = tile >> 4;           // batch
    const int n0   = (tile & 15) << 4;    // N tile origin
    const int lane = threadIdx.x;         // 0..31 (wave32)
    const int half = lane >> 4;           // selects K pair {0,1} vs {2,3}
    const int lp   = lane & 15;           // M (for A) / N (for B,C,D)
    const int slot = slots[b];
    const int row  = rows[b];
    const int n    = n0 + lp;

    // Loop-invariant: the scatter replacement value for column n.
    const float gval = table[row * DD + colid[n]];

    // Branchless A-row padding: clamp to a valid row, select 0 after the load.
    const bool  mvalid = (lp < AA);
    const float* qbase = q + (mvalid ? lp : 0) * (BB * DD) + b * DD;
    const float* bankb = bank + (b << 16) + n;   // + c*DD
    const float* emab  = ema  + (b << 16) + n;

    v8f acc = {};
#pragma unroll 4
    for (int k0 = 0; k0 < DD; k0 += 4) {
        const int kA = k0 + (half << 1);     // this lane's first K of the pair

        // A fragment: 16x4 f32 layout — lane lp holds A[M=lp, K=kA..kA+1]
        const float q0 = qbase[kA];
        const float q1 = qbase[kA + 1];
        v2f afrag;
        afrag.x = mvalid ? q0 : 0.0f;
        afrag.y = mvalid ? q1 : 0.0f;

        // B fragment: 4x16 f32 layout — lane lp holds B[K=kA..kA+1, N=n]
        const float b0 = bankb[(kA    ) << 8];
        const float b1 = bankb[(kA + 1) << 8];
        const float e0 = emab [(kA    ) << 8];
        const float e1 = emab [(kA + 1) << 8];
        v2f bfrag;
        bfrag.x = fmaf(e0, 0.975f, (kA     == slot) ? gval : b0);
        bfrag.y = fmaf(e1, 0.975f, (kA + 1 == slot) ? gval : b1);

        // (neg_a, A, neg_b, B, c_mod, C, reuse_a, reuse_b)
        acc = __builtin_amdgcn_wmma_f32_16x16x4_f32(false, afrag, false, bfrag,
                                                    (short)0, acc, false, false);
    }

    // C/D 16x16 f32 layout: VGPR r holds M = r + half*8, N = n
#pragma unroll
    for (int r = 0; r < 8; ++r) {
        const int m = r + (half << 3);
        if (m < AA)
            out1[m * (BB * DD) + b * DD + n] = acc[r];
    }
}

// out2[b,d,c] = add[b,c,d] — LDS-tiled transpose (32x33 padded tile, conflict-free),
// scatter + EMA fused on the load side, branchless select.
__global__ __launch_bounds__(256)
void out2_transpose_kernel(const float* __restrict__ bank,
                           const float* __restrict__ table,
                           const int*   __restrict__ rows,
                           const int*   __restrict__ colid,
                           const int*   __restrict__ slots,
                           const float* __restrict__ ema,
                           float*       __restrict__ out2) {  // [6,256,256]
    __shared__ float tile[32][33];
    const int b  = blockIdx.z;
    const int c0 = blockIdx.y << 5;
    const int d0 = blockIdx.x << 5;
    const int tx = threadIdx.x;   // 0..31
    const int ty = threadIdx.y;   // 0..7
    const int slot = slots[b];
    const int row  = rows[b];
    const int d    = d0 + tx;

    // Loop-invariant scatter replacement value for column d.
    const float gval = table[row * DD + colid[d]];

    const float* bankb = bank + (b << 16) + d;
    const float* emab  = ema  + (b << 16) + d;

#pragma unroll
    for (int i = 0; i < 4; ++i) {
        const int c  = c0 + ty + (i << 3);
        const float bval = bankb[c << 8];
        const float eval = emab [c << 8];
        tile[ty + (i << 3)][tx] = fmaf(eval, 0.975f, (c == slot) ? gval : bval);
    }
    __syncthreads();
#pragma unroll
    for (int i = 0; i < 4; ++i) {
        const int dd = d0 + ty + (i << 3);
        out2[(b << 16) + (dd << 8) + (c0 + tx)] = tile[tx][ty + (i << 3)];
    }
}

extern "C" void kernel_launch(void* const* d_in, const int* in_sizes, int n_in,
                              void* d_out, int out_size, void* d_ws, size_t ws_size,
                              hipStream_t stream) {
    const float* p1 = (const float*)d_in[0];   // bank   [6,256,256]
    const float* p2 = (const float*)d_in[1];   // table  [131072,256]
    const int*   p3 = (const int*)  d_in[2];   // rows   [6]
    const int*   p4 = (const int*)  d_in[3];   // colid  [256]
    const int*   p5 = (const int*)  d_in[4];   // slots  [6]
    // d_in[5] = primals_6 = arange(256): identity column scatter, not needed
    const float* p7 = (const float*)d_in[6];   // ema    [6,256,256]
    const float* p8 = (const float*)d_in[7];   // q      [12,6,256]

    float* out1 = (float*)d_out;               // 12*6*256 = 18432 floats
    float* out2 = out1 + AA * BB * DD;         // 6*256*256 floats

    out1_wmma_kernel<<<dim3(96), dim3(32), 0, stream>>>(p1, p2, p3, p4, p5, p7, p8, out1);

    dim3 tgrid(DD / 32, SS / 32, BB);          // (8,8,6)
    dim3 tblk(32, 8);
    out2_transpose_kernel<<<tgrid, tblk, 0, stream>>>(p1, p2, p3, p4, p5, p7, out2);
}